// MRConv1d_80358838108753
// MI455X (gfx1250) — compile-verified
//
#include <hip/hip_runtime.h>

// MRConv (max-relative graph conv) fused kernel for gfx1250.
//  out = relu( [x | max_k(x_j - x_i)] @ W + b )
// N=100000, C=128, K=16, C_OUT=128. Memory-bound on the neighbor gather
// (~819 MB logical, L2-resident since x is 51 MB < 192 MB L2); GEMM uses
// full-precision V_WMMA_F32_16X16X4_F32 (compute is only 6.6 GFLOP).

typedef __attribute__((ext_vector_type(2))) float v2f;
typedef __attribute__((ext_vector_type(4))) float v4f;
typedef __attribute__((ext_vector_type(8))) float v8f;

#define C_IN   128
#define C_OUTD 128
#define KNN    16
#define TILE_M 16
// Row stride 260 floats: row-to-row LDS bank offset = 260 % 64 = 4 banks, so
// the 16 rows of an A-fragment ds_load_b64 hit 16 distinct bank pairs (no
// conflicts). 260*4 = 1040 B is 16B-aligned -> float4 LDS stores are legal.
#define H_STRIDE 260

__global__ __launch_bounds__(128) void mrconv_wmma_kernel(
    const float* __restrict__ x,          // [N, 128]
    const long long* __restrict__ eidx,   // [2, N*16] int64; row 0 used
    const float* __restrict__ W,          // [256, 128]
    const float* __restrict__ bias,       // [128]
    float* __restrict__ out,              // [N, 128]
    int n_nodes)
{
  __shared__ float h[TILE_M][H_STRIDE];   // 16 x 260 x 4B = 16.6 KB

  const int tid  = threadIdx.x;                                  // 0..127
  const int base = blockIdx.x * TILE_M;
  // Wave id as a compiler-provable scalar so neighbor-index loads stay s_load.
  const int wv   = __builtin_amdgcn_readfirstlane(tid >> 5);     // 0..3
  const int ln   = tid & 31;
  const int cg   = ln << 2;                                      // 4 channels/lane

  // ---- Phase A: h = [x_i | max_k(x_j - x_i)], 4 rows in flight (1/wave) ----
  for (int r = 0; r < TILE_M / 4; ++r) {
    const int row  = (r << 2) + wv;
    const int node = base + row;
    // Uniform row base + divergent 32-bit lane offset -> GVS (saddr) loads:
    // scalar SALU computes the base, no per-neighbor v_add_nc_u64.
    const float* xrow = x + (long long)node * C_IN;
    const v4f xi = *(const v4f*)(xrow + cg);                        // b128
    const long long* ei = eidx + (long long)node * KNN;             // s_load_b512 x2
    v4f m = {-3.402823466e+38f, -3.402823466e+38f,
             -3.402823466e+38f, -3.402823466e+38f};
    #pragma unroll
    for (int k = 0; k < KNN; ++k) {
      const float* jrow = x + ei[k] * (long long)C_IN;              // uniform base
      const v4f xj = *(const v4f*)(jrow + cg);                      // b128 gather, saddr form
      m.x = fmaxf(m.x, xj.x - xi.x);
      m.y = fmaxf(m.y, xj.y - xi.y);
      m.z = fmaxf(m.z, xj.z - xi.z);
      m.w = fmaxf(m.w, xj.w - xi.w);
    }
    *(v4f*)(&h[row][cg])        = xi;   // ds_store_b128
    *(v4f*)(&h[row][C_IN + cg]) = m;    // ds_store_b128
  }
  __syncthreads();

  // ---- Phase B: C[16x128] = h[16x256] * W[256x128] via WMMA f32 16x16x4 ----
  const int half = ln >> 4;     // ISA A/B layout: lanes 0-15 hold K={0,1}, 16-31 K={2,3}
  const int l16  = ln & 15;
  const int n0   = wv * 32;     // each wave owns 32 output columns (2 N-tiles)

  v8f c0 = {};
  v8f c1 = {};
  #pragma unroll 4
  for (int k = 0; k < 2 * C_IN; k += 4) {
    const int kb = k + (half << 1);

    // A fragment: row l16, K = kb, kb+1 (consecutive -> ds_load_b64, conflict-free)
    v2f a;
    a.x = h[l16][kb];
    a.y = h[l16][kb + 1];

    // B fragments: W[kb..kb+1][n0+l16] and the +16 column tile (L2/WGP$-resident)
    const float* wp = W + (long long)kb * C_OUTD + n0 + l16;
    v2f b0, b1;
    b0.x = wp[0];            b0.y = wp[C_OUTD];
    b1.x = wp[16];           b1.y = wp[C_OUTD + 16];

    c0 = __builtin_amdgcn_wmma_f32_16x16x4_f32(false, a, false, b0,
                                               (short)0, c0, false, false);
    c1 = __builtin_amdgcn_wmma_f32_16x16x4_f32(false, a, false, b1,
                                               (short)0, c1, false, false);
  }

  // ---- Epilogue: bias + ReLU + store ----
  // C layout: VGPR v holds row M=v (lanes 0-15) / M=v+8 (lanes 16-31), N=l16.
  const float bias0 = bias[n0 + l16];
  const float bias1 = bias[n0 + 16 + l16];
  #pragma unroll
  for (int v = 0; v < 8; ++v) {
    const int m = v + (half << 3);
    float* op = out + (long long)(base + m) * C_OUTD + n0 + l16;
    op[0]  = fmaxf(c0[v] + bias0, 0.0f);
    op[16] = fmaxf(c1[v] + bias1, 0.0f);
  }
}

extern "C" void kernel_launch(void* const* d_in, const int* in_sizes, int n_in,
                              void* d_out, int out_size, void* d_ws, size_t ws_size,
                              hipStream_t stream) {
  const float*     x    = (const float*)d_in[0];
  const long long* eidx = (const long long*)d_in[1];   // int64 edge_index
  const float*     W    = (const float*)d_in[2];
  const float*     bias = (const float*)d_in[3];
  float*           out  = (float*)d_out;

  const int n_nodes = in_sizes[0] / C_IN;              // 100000
  const int blocks  = n_nodes / TILE_M;                // 6250 (exact)

  mrconv_wmma_kernel<<<blocks, 128, 0, stream>>>(x, eidx, W, bias, out, n_nodes);
}